// SplitBruteForceAntisymmetryWithDecay_54898271977534
// MI455X (gfx1250) — compile-verified
//
#include <hip/hip_runtime.h>
#include <math.h>

// ---------------- problem constants (match reference) ----------------
#define NELEC   12
#define NIONS   4
#define DBF     32      // backflow width d'
#define ND      32      // resnet hidden width
#define NSPLIT  2
#define NPER    6       // electrons per split
#define FLATW   192     // NPER * DBF
#define NPERM   720     // 6!
#define NTILE   45      // 720 / 16 perms per WMMA tile
#define BLOCK   128     // 4 wave32s
#define NWAVE   4

typedef __attribute__((ext_vector_type(2))) float v2f;
typedef __attribute__((ext_vector_type(8))) float v8f;

// Branch-free fp32 tanh: copysign(1 - 2/(e^{2|x|}+1), x).
// v_exp_f32 + v_rcp_f32 (~1 ulp each); saturates to +/-1 for large |x|.
__device__ __forceinline__ float fast_tanhf(float x) {
  float ax = fabsf(x);
  float e  = __expf(ax + ax);
  float t  = 1.0f - 2.0f * __builtin_amdgcn_rcpf(e + 1.0f);
  return copysignf(t, x);
}

// ---------------- perm/sign table via Lehmer code (lexicographic) ------------
__global__ void init_perms_kernel(unsigned char* __restrict__ perms,
                                  float* __restrict__ signs) {
  int p = blockIdx.x * blockDim.x + threadIdx.x;
  if (p >= NPERM) return;
  const int fact[6] = {120, 24, 6, 2, 1, 1};
  int avail[6] = {0, 1, 2, 3, 4, 5};
  int rem = p, nav = 6, par = 0;
  for (int i = 0; i < 6; ++i) {
    int d = rem / fact[i];
    rem -= d * fact[i];
    par += d;                       // Lehmer digit sum == inversion count
    perms[p * 6 + i] = (unsigned char)avail[d];
    for (int k = d; k < nav - 1; ++k) avail[k] = avail[k + 1];
    --nav;
  }
  signs[p] = (par & 1) ? -1.0f : 1.0f;
}

// ---------------- main kernel: one workgroup per walker ----------------------
__global__ __launch_bounds__(BLOCK) void ferminet_logpsi_kernel(
    const float* __restrict__ elec_pos,   // (B,12,3)
    const float* __restrict__ ion_pos,    // (4,3)
    const float* __restrict__ w_bf,       // (16,32)
    const float* __restrict__ b_bf,       // (32)
    const float* __restrict__ rw0,        // (2,192,32)
    const float* __restrict__ rb0,        // (2,32)
    const float* __restrict__ rw1,        // (2,32,32)
    const float* __restrict__ rb1,        // (2,32)
    const float* __restrict__ rwo,        // (2,32,1)
    const float* __restrict__ rbo,        // (2,1)
    const float* __restrict__ rates,      // (4)
    const unsigned char* __restrict__ perms_g, // (720,6)
    const float* __restrict__ signs_g,    // (720)
    float* __restrict__ out)              // (B)
{
  __shared__ float s_stream[NELEC][DBF];          // tanh backflow streams
  __shared__ float s_Y[NSPLIT][NPER][NPER][ND];   // Y[e][j] = x[e] @ rw0_blk[j]
  __shared__ float s_rw1[NSPLIT][ND * ND];
  __shared__ float s_rb0[NSPLIT][ND];
  __shared__ float s_rb1[NSPLIT][ND];
  __shared__ float s_rwo[NSPLIT][ND];
  __shared__ float s_rbo[NSPLIT];
  __shared__ float s_wbf[16 * DBF];
  __shared__ float s_bbf[DBF];
  __shared__ float s_ion[NIONS][3];
  __shared__ float s_rates[NIONS];
  __shared__ unsigned char s_perms[NPERM * 6];
  __shared__ float s_signs[NPERM];
  __shared__ float s_outs[NPERM];
  __shared__ float s_H0[NWAVE][16][ND];           // per-wave perm tile, pre-residual
  __shared__ float s_red[BLOCK];
  __shared__ float s_misc[4];                     // [0]=jastrow sum, [1..2]=log terms

  const int tid  = threadIdx.x;
  const int lane = tid & 31;
  const int wave = tid >> 5;
  const int b    = blockIdx.x;

  // ---- stage all weights / tables into LDS ----
  for (int o = tid; o < 16 * DBF; o += BLOCK) s_wbf[o] = w_bf[o];
  for (int o = tid; o < DBF; o += BLOCK)      s_bbf[o] = b_bf[o];
  for (int o = tid; o < NSPLIT * ND; o += BLOCK) {
    s_rb0[o >> 5][o & 31] = rb0[o];
    s_rb1[o >> 5][o & 31] = rb1[o];
    s_rwo[o >> 5][o & 31] = rwo[o];               // (2,32,1) flat == (2,32)
  }
  for (int o = tid; o < NSPLIT; o += BLOCK)   s_rbo[o] = rbo[o];
  for (int o = tid; o < NSPLIT * ND * ND; o += BLOCK)
    s_rw1[o >> 10][o & 1023] = rw1[o];            // [s][k*32+n]
  for (int o = tid; o < NIONS * 3; o += BLOCK) s_ion[o / 3][o % 3] = ion_pos[o];
  for (int o = tid; o < NIONS; o += BLOCK)     s_rates[o] = rates[o];
  for (int o = tid; o < NPERM; o += BLOCK)     s_signs[o] = signs_g[o];
  for (int o = tid; o < NPERM * 6; o += BLOCK) s_perms[o] = perms_g[o];
  __syncthreads();

  // ---- phase 1: backflow streams + per-electron jastrow partials ----
  for (int o = tid; o < NELEC * DBF; o += BLOCK) {
    int n = o >> 5, c = o & 31;
    const float* ep = &elec_pos[(b * NELEC + n) * 3];
    float ex = ep[0], ey = ep[1], ez = ep[2];
    float acc = s_bbf[c];
#pragma unroll
    for (int i = 0; i < NIONS; ++i) {
      float rx = ex - s_ion[i][0];
      float ry = ey - s_ion[i][1];
      float rz = ez - s_ion[i][2];
      float d = sqrtf(rx * rx + ry * ry + rz * rz);
      acc += rx * s_wbf[(i * 3 + 0) * 32 + c];
      acc += ry * s_wbf[(i * 3 + 1) * 32 + c];
      acc += rz * s_wbf[(i * 3 + 2) * 32 + c];
      acc += d * s_wbf[(12 + i) * 32 + c];
    }
    s_stream[n][c] = fast_tanhf(acc);
  }
  if (tid < NELEC) {
    const float* ep = &elec_pos[(b * NELEC + tid) * 3];
    float ex = ep[0], ey = ep[1], ez = ep[2];
    float jn = 0.0f;
#pragma unroll
    for (int i = 0; i < NIONS; ++i) {
      float rx = ex - s_ion[i][0];
      float ry = ey - s_ion[i][1];
      float rz = ez - s_ion[i][2];
      jn += sqrtf(rx * rx + ry * ry + rz * rz) * s_rates[i];
    }
    s_red[tid] = jn;
  }
  __syncthreads();
  if (tid == 0) {
    float j = 0.0f;
    for (int n = 0; n < NELEC; ++n) j += s_red[n];
    s_misc[0] = j;                                // log_jastrow = -j
  }

  // ---- phase 2: Y[s][e][j] = stream[s*6+e] @ rw0[s][j*32:(j+1)*32] ----
  for (int o = tid; o < NSPLIT * NPER * NPER * ND; o += BLOCK) {
    int s = o / (NPER * NPER * ND);
    int rem = o - s * (NPER * NPER * ND);
    int e = rem / (NPER * ND);   rem -= e * (NPER * ND);
    int j = rem >> 5;
    int n = rem & 31;
    const float* w = &rw0[(s * FLATW + j * 32) * 32 + n];
    float acc = 0.0f;
#pragma unroll 8
    for (int k = 0; k < 32; ++k) acc += s_stream[s * NPER + e][k] * w[k * 32];
    s_Y[s][e][j][n] = acc;
  }
  __syncthreads();

  // ---- phase 3: per split, 45 tiles of 16 perms; residual GEMM on WMMA ----
  const int hi = lane >> 4;       // half-wave: 0 or 1
  const int nn = lane & 15;       // M index for A frags / N index for B frags

  for (int s = 0; s < NSPLIT; ++s) {
    for (int t = wave; t < NTILE; t += NWAVE) {
      // build H0 tile: h0[p] = tanh(rb0 + sum_j Y[perm_p[j], j]); lane = channel
#pragma unroll
      for (int it = 0; it < 16; ++it) {
        int p = t * 16 + it;
        const unsigned char* pp = &s_perms[p * 6];
        float acc = s_rb0[s][lane];
#pragma unroll
        for (int j = 0; j < 6; ++j) acc += s_Y[s][pp[j]][j][lane];
        s_H0[wave][it][lane] = fast_tanhf(acc);
      }
      // DS ops from one wave complete in order -> safe to read back below.

      // H0(16x32) @ rw1(32x32): 8 k-steps of v_wmma_f32_16x16x4_f32, 2 N-tiles,
      // 4 independent accumulator chains for ILP.
      v8f d0a = {0.f, 0.f, 0.f, 0.f, 0.f, 0.f, 0.f, 0.f};
      v8f d0b = d0a, d1a = d0a, d1b = d0a;
#pragma unroll
      for (int kk = 0; kk < 4; ++kk) {
        int k0 = kk * 4 + 2 * hi;       // A/B layout: VGPR0 K=k0, VGPR1 K=k0+1
        int k1 = k0 + 16;
        v2f a0, a1, b00, b01, b10, b11;
        a0[0] = s_H0[wave][nn][k0];
        a0[1] = s_H0[wave][nn][k0 + 1];
        a1[0] = s_H0[wave][nn][k1];
        a1[1] = s_H0[wave][nn][k1 + 1];
        b00[0] = s_rw1[s][k0 * 32 + nn];
        b00[1] = s_rw1[s][(k0 + 1) * 32 + nn];
        b01[0] = s_rw1[s][k0 * 32 + nn + 16];
        b01[1] = s_rw1[s][(k0 + 1) * 32 + nn + 16];
        b10[0] = s_rw1[s][k1 * 32 + nn];
        b10[1] = s_rw1[s][(k1 + 1) * 32 + nn];
        b11[0] = s_rw1[s][k1 * 32 + nn + 16];
        b11[1] = s_rw1[s][(k1 + 1) * 32 + nn + 16];
        d0a = __builtin_amdgcn_wmma_f32_16x16x4_f32(false, a0, false, b00,
                                                    (short)0, d0a, false, false);
        d1a = __builtin_amdgcn_wmma_f32_16x16x4_f32(false, a0, false, b01,
                                                    (short)0, d1a, false, false);
        d0b = __builtin_amdgcn_wmma_f32_16x16x4_f32(false, a1, false, b10,
                                                    (short)0, d0b, false, false);
        d1b = __builtin_amdgcn_wmma_f32_16x16x4_f32(false, a1, false, b11,
                                                    (short)0, d1b, false, false);
      }
      v8f d0 = d0a + d0b;
      v8f d1 = d1a + d1b;

      // epilogue in registers: H = tanh(D + rb1) + H0, then head dot via
      // cross-lane reduction inside each 16-lane half (D row = r + 8*hi,
      // lane nn holds columns nn and nn+16 of that row).
      float w0 = s_rwo[s][nn];
      float w1 = s_rwo[s][nn + 16];
#pragma unroll
      for (int r = 0; r < 8; ++r) {
        int M = r + 8 * hi;
        float ha = fast_tanhf(d0[r] + s_rb1[s][nn])      + s_H0[wave][M][nn];
        float hb = fast_tanhf(d1[r] + s_rb1[s][nn + 16]) + s_H0[wave][M][nn + 16];
        float part = ha * w0 + hb * w1;
        part += __shfl_xor(part, 1, 32);
        part += __shfl_xor(part, 2, 32);
        part += __shfl_xor(part, 4, 32);
        part += __shfl_xor(part, 8, 32);      // masks < 16: stays in half-wave
        if (nn == r) {
          int p = t * 16 + M;
          s_outs[p] = s_signs[p] * (part + s_rbo[s]);
        }
      }
    }
    __syncthreads();

    // ---- stabilized log|sum| over 720 signed values ----
    float lm = 0.0f;
    for (int p = tid; p < NPERM; p += BLOCK) lm = fmaxf(lm, fabsf(s_outs[p]));
    s_red[tid] = lm;
    __syncthreads();
    for (int off = 64; off > 0; off >>= 1) {
      if (tid < off) s_red[tid] = fmaxf(s_red[tid], s_red[tid + off]);
      __syncthreads();
    }
    float mmax = s_red[0];
    __syncthreads();

    float sm = 0.0f;
    for (int p = tid; p < NPERM; p += BLOCK) sm += s_outs[p] / mmax;
    s_red[tid] = sm;
    __syncthreads();
    for (int off = 64; off > 0; off >>= 1) {
      if (tid < off) s_red[tid] += s_red[tid + off];
      __syncthreads();
    }
    if (tid == 0) s_misc[1 + s] = logf(fabsf(s_red[0])) + logf(mmax);
    __syncthreads();
  }

  if (tid == 0) out[b] = s_misc[1] + s_misc[2] - s_misc[0];
}

// ---------------- launch ----------------
extern "C" void kernel_launch(void* const* d_in, const int* in_sizes, int n_in,
                              void* d_out, int out_size, void* d_ws, size_t ws_size,
                              hipStream_t stream) {
  const float* elec_pos = (const float*)d_in[0];
  const float* ion_pos  = (const float*)d_in[1];
  const float* w_bf     = (const float*)d_in[2];
  const float* b_bf     = (const float*)d_in[3];
  const float* rw0      = (const float*)d_in[4];
  const float* rb0      = (const float*)d_in[5];
  const float* rw1      = (const float*)d_in[6];
  const float* rb1      = (const float*)d_in[7];
  const float* rwo      = (const float*)d_in[8];
  const float* rbo      = (const float*)d_in[9];
  const float* rates    = (const float*)d_in[10];
  float* out = (float*)d_out;

  // workspace: [0, 2880) signs f32[720]; [2880, 7200) perms u8[720*6]
  float* signs = (float*)d_ws;
  unsigned char* perms = (unsigned char*)d_ws + NPERM * sizeof(float);

  int B = in_sizes[0] / (NELEC * 3);

  init_perms_kernel<<<(NPERM + 127) / 128, 128, 0, stream>>>(perms, signs);
  ferminet_logpsi_kernel<<<B, BLOCK, 0, stream>>>(
      elec_pos, ion_pos, w_bf, b_bf, rw0, rb0, rw1, rb1, rwo, rbo, rates,
      perms, signs, out);
}